// HSTGNN_29308856828503
// MI455X (gfx1250) — compile-verified
//
#include <hip/hip_runtime.h>

#define HD 96
#define BD 32
#define ICD 8
#define OCD 2
#define GEMM_MAXK 192
#define GEMM_LDSROW (GEMM_MAXK + 2)

typedef __attribute__((ext_vector_type(2))) float v2f;
typedef __attribute__((ext_vector_type(8))) float v8f;

__device__ __forceinline__ float gelu_f(float x) {
  return 0.5f * x * (1.0f + erff(x * 0.70710678118654752440f));
}
__device__ __forceinline__ float sigm_f(float x) {
  return 1.0f / (1.0f + __expf(-x));
}
// monotonic float <-> uint mapping for atomicMax-based segment max
__device__ __forceinline__ unsigned f2ord(float f) {
  unsigned u = __float_as_uint(f);
  return (u & 0x80000000u) ? ~u : (u | 0x80000000u);
}
__device__ __forceinline__ float ord2f(unsigned u) {
  return (u & 0x80000000u) ? __uint_as_float(u & 0x7fffffffu)
                           : __uint_as_float(~u);
}

// ------------------------------------------------------------------
// Dense GEMM via fp32 WMMA (V_WMMA_F32_16X16X4_F32).
// One wave per 16x16 output tile; 4 waves (one block) share a column
// tile whose W[K x 16] slab is staged transposed in LDS (row stride
// K+2 -> bank-conflict-free b64 reads).  A-fragment loads are
// unconditional global_load_b64 on the full-tile fast path.
// D = act(A[n,K] * W[K,M] + bias + C0); act: 0=none, 1=GELU, 2=sigmoid
// ------------------------------------------------------------------
__global__ __launch_bounds__(128) void gemm_bias_act(
    const float* __restrict__ A, const float* __restrict__ W,
    const float* __restrict__ bias, const float* __restrict__ C0, int ldC0,
    float* __restrict__ D, int ldD, int n, int K, int M, int act) {
  __shared__ float wt[16 * GEMM_LDSROW];
  const int lane = threadIdx.x & 31;
  const int wv = threadIdx.x >> 5;
  const int tilesM = (M + 15) >> 4;
  const int ct = blockIdx.x % tilesM;
  const int rg = blockIdx.x / tilesM;
  const int col0 = ct * 16;
  const int KP = K + 2;

  // cooperative staging of W[:, col0:col0+16]^T (zero-pad ragged cols)
  for (int idx = threadIdx.x; idx < K * 16; idx += 128) {
    const int k = idx >> 4, c = idx & 15;
    const float v = (col0 + c < M) ? W[(size_t)k * M + col0 + c] : 0.f;
    wt[c * KP + k] = v;
  }
  __syncthreads();

  const int tr = rg * 4 + wv;
  const int row0 = tr * 16;
  if (row0 >= n) return;            // wave-uniform; after the barrier

  const int sub = lane & 15;
  const int koff = (lane >> 4) << 1;  // lanes 16-31 hold K+2,K+3
  const int ar = row0 + sub;
  const float* lb = &wt[sub * KP + koff];

  v8f acc = {0.f, 0.f, 0.f, 0.f, 0.f, 0.f, 0.f, 0.f};
  if (row0 + 16 <= n) {
    // fast path: branch-free, vector loads
    const float* ap = A + (size_t)ar * K + koff;
    for (int k0 = 0; k0 < K; k0 += 4) {
      const v2f a = *(const v2f*)(ap + k0);
      const v2f b = *(const v2f*)(lb + k0);
      acc = __builtin_amdgcn_wmma_f32_16x16x4_f32(false, a, false, b, (short)0,
                                                  acc, false, false);
    }
  } else {
    const bool arok = (ar < n);
    for (int k0 = 0; k0 < K; k0 += 4) {
      const int ka = k0 + koff;
      v2f a;
      a.x = arok ? A[(size_t)ar * K + ka] : 0.f;
      a.y = arok ? A[(size_t)ar * K + ka + 1] : 0.f;
      const v2f b = *(const v2f*)(lb + k0);
      acc = __builtin_amdgcn_wmma_f32_16x16x4_f32(false, a, false, b, (short)0,
                                                  acc, false, false);
    }
  }

  const int oc = col0 + sub;
  if (oc >= M) return;
  const int rbase = row0 + ((lane >> 4) << 3);
  const float bv = bias ? bias[oc] : 0.f;
#pragma unroll
  for (int i = 0; i < 8; ++i) {
    const int orow = rbase + i;
    if (orow < n) {
      float vv = acc[i] + bv;
      if (C0) vv += C0[(size_t)orow * ldC0 + oc];
      if (act == 1) vv = gelu_f(vv);
      else if (act == 2) vv = sigm_f(vv);
      D[(size_t)orow * ldD + oc] = vv;
    }
  }
}

// ------------------------------------------------------------------
// graph / elementwise kernels
// ------------------------------------------------------------------
__global__ void fill_zero_k(float* __restrict__ p, long long n) {
  long long i = (long long)blockIdx.x * blockDim.x + threadIdx.x;
  if (i < n) p[i] = 0.f;
}

__global__ void deg_cnt_k(const int* __restrict__ src, const int* __restrict__ dst,
                          float* __restrict__ deg, float* __restrict__ cnt, int E) {
  int e = blockIdx.x * blockDim.x + threadIdx.x;
  if (e < E) {
    atomicAdd(&deg[src[e]], 1.f);
    atomicAdd(&cnt[dst[e]], 1.f);
  }
}

__global__ void node_norms_k(const float* __restrict__ deg, float* __restrict__ cnt,
                             float* __restrict__ dis, int n) {
  int i = blockIdx.x * blockDim.x + threadIdx.x;
  if (i < n) {
    float d = deg[i];
    dis[i] = (d > 0.f) ? rsqrtf(fmaxf(d, 1.f)) : 0.f;
    cnt[i] = fmaxf(cnt[i], 1.f);
  }
}

__global__ void edge_cheb_k(const float* __restrict__ dis, const int* __restrict__ src,
                            const int* __restrict__ dst, float* __restrict__ cn, int E) {
  int e = blockIdx.x * blockDim.x + threadIdx.x;
  if (e < E) cn[e] = -(dis[src[e]] * dis[dst[e]]);
}

// out[dst[e], f] += w(e) * in[src[e], f]
__global__ void scatter_add_k(const float* __restrict__ in, const int* __restrict__ src,
                              const int* __restrict__ dst, const float* __restrict__ wE,
                              float* __restrict__ out, int E, int F) {
  long long idx = (long long)blockIdx.x * blockDim.x + threadIdx.x;
  if (idx >= (long long)E * F) return;
  int e = (int)(idx / F), f = (int)(idx % F);
  float w = wE ? wE[e] : 1.f;
  atomicAdd(&out[(size_t)dst[e] * F + f], w * in[(size_t)src[e] * F + f]);
}

__global__ void rowdiv_k(float* __restrict__ p, const float* __restrict__ cnt,
                         long long total, int F) {
  long long i = (long long)blockIdx.x * blockDim.x + threadIdx.x;
  if (i < total) p[i] /= cnt[i / F];
}

__global__ void cheb_t2_k(float* __restrict__ t2, const float* __restrict__ x,
                          long long n) {
  long long i = (long long)blockIdx.x * blockDim.x + threadIdx.x;
  if (i < n) t2[i] = 2.f * t2[i] - x[i];
}

__global__ void edge_logits_k(const float* __restrict__ q, const float* __restrict__ kk,
                              const int* __restrict__ src, const int* __restrict__ dst,
                              float* __restrict__ lg, int E, float scale) {
  int e = blockIdx.x * blockDim.x + threadIdx.x;
  if (e >= E) return;
  const float* qd = q + (size_t)dst[e] * BD;
  const float* ks = kk + (size_t)src[e] * BD;
  float s = 0.f;
#pragma unroll
  for (int i = 0; i < BD; ++i) s += qd[i] * ks[i];
  lg[e] = s * scale;
}

__global__ void seg_max_k(const float* __restrict__ lg, unsigned* __restrict__ mk,
                          const int* __restrict__ dst, int E) {
  int e = blockIdx.x * blockDim.x + threadIdx.x;
  if (e < E) atomicMax(&mk[dst[e]], f2ord(lg[e]));
}

__global__ void seg_expsum_k(const float* __restrict__ lg, const unsigned* __restrict__ mk,
                             const int* __restrict__ dst, float* __restrict__ ex,
                             float* __restrict__ s, int E) {
  int e = blockIdx.x * blockDim.x + threadIdx.x;
  if (e >= E) return;
  float m = ord2f(mk[dst[e]]);
  float v = __expf(lg[e] - m);
  ex[e] = v;
  atomicAdd(&s[dst[e]], v);
}

__global__ void edge_alpha_k(float* __restrict__ ex, const float* __restrict__ s,
                             const int* __restrict__ dst, int E) {
  int e = blockIdx.x * blockDim.x + threadIdx.x;
  if (e < E) ex[e] = ex[e] / (s[dst[e]] + 1e-16f);
}

// wave32 per node LayerNorm over 96 features (3 per lane + shfl tree)
__global__ __launch_bounds__(128) void layernorm96_k(
    const float* __restrict__ in, const float* __restrict__ gg,
    const float* __restrict__ bb, float* __restrict__ out, int n, int dogelu) {
  int node = blockIdx.x * 4 + (threadIdx.x >> 5);
  int lane = threadIdx.x & 31;
  if (node >= n) return;
  const float* r = in + (size_t)node * HD;
  float v0 = r[lane], v1 = r[lane + 32], v2 = r[lane + 64];
  float s = v0 + v1 + v2;
  for (int o = 16; o > 0; o >>= 1) s += __shfl_xor(s, o, 32);
  float mean = s * (1.f / HD);
  float d0 = v0 - mean, d1 = v1 - mean, d2 = v2 - mean;
  float q = d0 * d0 + d1 * d1 + d2 * d2;
  for (int o = 16; o > 0; o >>= 1) q += __shfl_xor(q, o, 32);
  float inv = rsqrtf(q * (1.f / HD) + 1e-5f);
  float* w = out + (size_t)node * HD;
  float r0 = d0 * inv * gg[lane] + bb[lane];
  float r1 = d1 * inv * gg[lane + 32] + bb[lane + 32];
  float r2 = d2 * inv * gg[lane + 64] + bb[lane + 64];
  if (dogelu) { r0 = gelu_f(r0); r1 = gelu_f(r1); r2 = gelu_f(r2); }
  w[lane] = r0; w[lane + 32] = r1; w[lane + 64] = r2;
}

__global__ void bn_inplace_k(float* __restrict__ h, const float* __restrict__ g,
                             const float* __restrict__ b, const float* __restrict__ m,
                             const float* __restrict__ v, long long n) {
  long long i = (long long)blockIdx.x * blockDim.x + threadIdx.x;
  if (i >= n) return;
  int c = (int)(i % HD);
  h[i] = (h[i] - m[c]) * rsqrtf(v[c] + 1e-5f) * g[c] + b[c];
}

__global__ void gate_combine_k(const float* __restrict__ gate, const float* __restrict__ cat,
                               const float* __restrict__ x, float* __restrict__ out,
                               long long n) {
  long long i = (long long)blockIdx.x * blockDim.x + threadIdx.x;
  if (i < n) out[i] = gate[i] * cat[i] + x[i];
}

__global__ void tscale_k(const float* __restrict__ x, const float* __restrict__ sc,
                         const float* __restrict__ sh, float* __restrict__ out,
                         long long n) {
  long long i = (long long)blockIdx.x * blockDim.x + threadIdx.x;
  if (i >= n) return;
  int c = (int)(i % HD);
  out[i] = x[i] * sc[c] + sh[c];
}

__global__ void tupdate_k(float* __restrict__ x, const float* __restrict__ xln,
                          const float* __restrict__ gate, long long n) {
  long long i = (long long)blockIdx.x * blockDim.x + threadIdx.x;
  if (i < n) x[i] += tanhf(gate[0]) * xln[i];
}

__global__ void gat_coef_k(const float* __restrict__ xp, const float* __restrict__ as,
                           const float* __restrict__ ad, float* __restrict__ osrc,
                           float* __restrict__ odst, int n) {
  int idx = blockIdx.x * blockDim.x + threadIdx.x;
  if (idx >= n * 4) return;
  int node = idx >> 2, h = idx & 3;
  const float* r = xp + (size_t)node * HD + h * 24;
  float s1 = 0.f, s2 = 0.f;
#pragma unroll
  for (int d = 0; d < 24; ++d) {
    float v = r[d];
    s1 += v * as[h * 24 + d];
    s2 += v * ad[h * 24 + d];
  }
  osrc[idx] = s1;
  odst[idx] = s2;
}

// src/dst == nullptr -> self-loop pass (edge e is node e)
__global__ void gat_logit_max_k(const float* __restrict__ asrc, const float* __restrict__ adst,
                                const int* __restrict__ src, const int* __restrict__ dst,
                                float* __restrict__ lg, unsigned* __restrict__ mk, int cnt) {
  int idx = blockIdx.x * blockDim.x + threadIdx.x;
  if (idx >= cnt * 4) return;
  int e = idx >> 2, h = idx & 3;
  int s_ = src ? src[e] : e, d_ = dst ? dst[e] : e;
  float v = asrc[s_ * 4 + h] + adst[d_ * 4 + h];
  v = (v > 0.f) ? v : 0.2f * v;   // leaky_relu 0.2
  lg[idx] = v;
  atomicMax(&mk[d_ * 4 + h], f2ord(v));
}

__global__ void gat_expsum_k(float* __restrict__ lg, const unsigned* __restrict__ mk,
                             const int* __restrict__ dst, float* __restrict__ s, int cnt) {
  int idx = blockIdx.x * blockDim.x + threadIdx.x;
  if (idx >= cnt * 4) return;
  int e = idx >> 2, h = idx & 3;
  int d_ = dst ? dst[e] : e;
  float m = ord2f(mk[d_ * 4 + h]);
  float v = __expf(lg[idx] - m);
  lg[idx] = v;
  atomicAdd(&s[d_ * 4 + h], v);
}

__global__ void gat_alpha_k(float* __restrict__ lg, const float* __restrict__ s,
                            const int* __restrict__ dst, int cnt) {
  int idx = blockIdx.x * blockDim.x + threadIdx.x;
  if (idx >= cnt * 4) return;
  int e = idx >> 2, h = idx & 3;
  int d_ = dst ? dst[e] : e;
  lg[idx] /= (s[d_ * 4 + h] + 1e-16f);
}

__global__ void gat_scatter_k(const float* __restrict__ xp, const int* __restrict__ src,
                              const int* __restrict__ dst, const float* __restrict__ al,
                              float* __restrict__ go, int cnt) {
  long long idx = (long long)blockIdx.x * blockDim.x + threadIdx.x;
  if (idx >= (long long)cnt * HD) return;
  int e = (int)(idx / HD), f = (int)(idx % HD);
  int h = f / 24;
  int s_ = src ? src[e] : e, d_ = dst ? dst[e] : e;
  atomicAdd(&go[(size_t)d_ * HD + f], al[e * 4 + h] * xp[(size_t)s_ * HD + f]);
}

__global__ void bn_gelu_add_k(const float* __restrict__ go, const float* __restrict__ gbias,
                              const float* __restrict__ g, const float* __restrict__ b,
                              const float* __restrict__ m, const float* __restrict__ v,
                              float* __restrict__ x, long long n) {
  long long i = (long long)blockIdx.x * blockDim.x + threadIdx.x;
  if (i >= n) return;
  int c = (int)(i % HD);
  float t = (go[i] + gbias[c] - m[c]) * rsqrtf(v[c] + 1e-5f) * g[c] + b[c];
  x[i] += gelu_f(t);
}

__global__ void final_mix_k(const float* __restrict__ mix, const float* __restrict__ gp,
                            const float* __restrict__ fp, const float* __restrict__ sk,
                            float* __restrict__ out, long long n) {
  long long i = (long long)blockIdx.x * blockDim.x + threadIdx.x;
  if (i >= n) return;
  float mv = mix[i];
  out[i] = mv * gp[i] + (1.f - mv) * fp[i] + sk[i];
}

// ------------------------------------------------------------------
// host orchestration
// ------------------------------------------------------------------
extern "C" void kernel_launch(void* const* d_in, const int* in_sizes, int n_in,
                              void* d_out, int out_size, void* d_ws, size_t ws_size,
                              hipStream_t stream) {
  (void)n_in; (void)out_size; (void)ws_size;
  const int N = in_sizes[0] / ICD;
  const int E = in_sizes[1] / 2;

  int ix = 0;
  auto nx = [&]() { return (const float*)d_in[ix++]; };
  const float* X = nx();
  const int* EI = (const int*)d_in[ix++];
  const int* src = EI;
  const int* dst = EI + E;

  // inp
  const float *inp_w = nx(), *inp_b = nx(), *ibg = nx(), *ibb = nx(), *ibm = nx(), *ibv = nx();
  // blocks
  struct Blk { const float *lw,*lb,*rw,*cw,*cb,*qw,*qb,*kw,*kb,*vw,*vb,*sw,*sb,*gw,*gb,*lng,*lnb; };
  Blk blk[3];
  for (int i = 0; i < 3; ++i) {
    blk[i].lw = nx(); blk[i].lb = nx(); blk[i].rw = nx();
    blk[i].cw = nx(); blk[i].cb = nx();
    blk[i].qw = nx(); blk[i].qb = nx(); blk[i].kw = nx(); blk[i].kb = nx();
    blk[i].vw = nx(); blk[i].vb = nx(); blk[i].sw = nx(); blk[i].sb = nx();
    blk[i].gw = nx(); blk[i].gb = nx(); blk[i].lng = nx(); blk[i].lnb = nx();
  }
  // temporal
  struct Tmp { const float *scale,*shift,*gate,*lng,*lnb,*w1,*b1,*w2,*b2; };
  Tmp tp[2];
  for (int i = 0; i < 2; ++i) {
    tp[i].scale = nx(); tp[i].shift = nx(); tp[i].gate = nx();
    tp[i].lng = nx(); tp[i].lnb = nx();
    tp[i].w1 = nx(); tp[i].b1 = nx(); tp[i].w2 = nx(); tp[i].b2 = nx();
  }
  // gat + bn
  const float *gat_w = nx(), *gat_as = nx(), *gat_ad = nx(), *gat_bias = nx();
  const float *bng = nx(), *bnb = nx(), *bnm = nx(), *bnv = nx();
  // feat
  const float *fw1 = nx(), *fb1 = nx(), *flng = nx(), *flnb = nx(), *fw2 = nx(), *fb2 = nx();
  // heads
  const float *g1w = nx(), *g1b = nx(), *g2w = nx(), *g2b = nx(), *g3w = nx(), *g3b = nx();
  const float *f1w = nx(), *f1b = nx(), *f2w = nx(), *f2b = nx();
  const float *m1w = nx(), *m1b = nx(), *m2w = nx(), *m2b = nx();
  const float *skw = nx(), *skb = nx();

  // workspace carve-out
  float* wsf = (float*)d_ws;
  size_t off = 0;
  auto alloc = [&](size_t nfl) { float* p = wsf + off; off += nfl; return p; };
  float *deg = alloc(N), *cnt = alloc(N), *dis = alloc(N);
  float* cheb = alloc(E);
  float *hA = alloc((size_t)N * HD), *hB = alloc((size_t)N * HD);
  float* nbuf[6];
  for (int i = 0; i < 6; ++i) nbuf[i] = alloc((size_t)N * HD);
  float* nb192 = alloc((size_t)N * 2 * HD);
  float *t32 = alloc((size_t)N * BD), *qb = alloc((size_t)N * BD),
        *kb = alloc((size_t)N * BD), *vb = alloc((size_t)N * BD);
  float *elog = alloc(E), *ealpha = alloc(E);
  float *segm = alloc(N), *segs = alloc(N);
  float *asrc = alloc((size_t)N * 4), *adst = alloc((size_t)N * 4);
  float *mkey4 = alloc((size_t)N * 4), *s4 = alloc((size_t)N * 4);
  float* lgN = alloc((size_t)N * 4);
  float* lgE = alloc((size_t)E * 4);
  float *gp = alloc((size_t)N * OCD), *fp = alloc((size_t)N * OCD),
        *mixv = alloc((size_t)N * OCD), *skv = alloc((size_t)N * OCD);
  float* h48a = nb192;                    // N*48
  float* h48b = nb192 + (size_t)N * 48;   // N*48

  auto gb = [](long long n) { return (unsigned)((n + 255) / 256); };
  auto zero = [&](float* p, long long n) {
    fill_zero_k<<<gb(n), 256, 0, stream>>>(p, n);
  };
  auto gemm = [&](const float* A, const float* W, const float* bias,
                  const float* C0, int ldC0, float* D, int ldD, int n, int K,
                  int M, int act) {
    int tilesM = (M + 15) / 16;
    int rowGroups = ((n + 15) / 16 + 3) / 4;
    gemm_bias_act<<<tilesM * rowGroups, 128, 0, stream>>>(A, W, bias, C0, ldC0,
                                                          D, ldD, n, K, M, act);
  };
  auto lnorm = [&](const float* in, const float* g, const float* b, float* out,
                   int gel) {
    layernorm96_k<<<(N + 3) / 4, 128, 0, stream>>>(in, g, b, out, N, gel);
  };
  const long long NH = (long long)N * HD;

  // ---- graph normalization precompute ----
  zero(deg, N); zero(cnt, N);
  deg_cnt_k<<<gb(E), 256, 0, stream>>>(src, dst, deg, cnt, E);
  node_norms_k<<<gb(N), 256, 0, stream>>>(deg, cnt, dis, N);
  edge_cheb_k<<<gb(E), 256, 0, stream>>>(dis, src, dst, cheb, E);

  // ---- input layer: bn(gelu(x@W+b)) ----
  gemm(X, inp_w, inp_b, nullptr, 0, hA, HD, N, ICD, HD, 1);
  bn_inplace_k<<<gb(NH), 256, 0, stream>>>(hA, ibg, ibb, ibm, ibv, NH);

  float* hin = hA;
  float* hout = hB;
  for (int bi = 0; bi < 3; ++bi) {
    const Blk& P = blk[bi];
    float *mean = nbuf[0], *t1 = nbuf[1], *t2 = nbuf[2], *cat = nbuf[3],
          *gatebuf = nbuf[4], *comb = nbuf[5];
    // SAGE: xl = gelu(mean@lw + lb + x@rw)  -> cat[:, 0:32]
    zero(mean, NH);
    scatter_add_k<<<gb((long long)E * HD), 256, 0, stream>>>(hin, src, dst, nullptr, mean, E, HD);
    rowdiv_k<<<gb(NH), 256, 0, stream>>>(mean, cnt, NH, HD);
    gemm(hin, P.rw, nullptr, nullptr, 0, t32, BD, N, HD, BD, 0);
    gemm(mean, P.lw, P.lb, t32, BD, cat + 0, HD, N, HD, BD, 1);
    // Cheb K=3: t1 = L x, t2 = 2 L t1 - x; xs -> cat[:, 32:64]
    zero(t1, NH);
    scatter_add_k<<<gb((long long)E * HD), 256, 0, stream>>>(hin, src, dst, cheb, t1, E, HD);
    zero(t2, NH);
    scatter_add_k<<<gb((long long)E * HD), 256, 0, stream>>>(t1, src, dst, cheb, t2, E, HD);
    cheb_t2_k<<<gb(NH), 256, 0, stream>>>(t2, hin, NH);
    gemm(hin, P.cw, nullptr, nullptr, 0, t32, BD, N, HD, BD, 0);
    gemm(t1, P.cw + HD * BD, nullptr, t32, BD, t32, BD, N, HD, BD, 0);
    gemm(t2, P.cw + 2 * HD * BD, P.cb, t32, BD, cat + BD, HD, N, HD, BD, 1);
    // TransformerConv: xa -> cat[:, 64:96]
    gemm(hin, P.qw, P.qb, nullptr, 0, qb, BD, N, HD, BD, 0);
    gemm(hin, P.kw, P.kb, nullptr, 0, kb, BD, N, HD, BD, 0);
    gemm(hin, P.vw, P.vb, nullptr, 0, vb, BD, N, HD, BD, 0);
    edge_logits_k<<<gb(E), 256, 0, stream>>>(qb, kb, src, dst, elog, E, 0.17677669529663689f);
    zero(segm, N); zero(segs, N);
    seg_max_k<<<gb(E), 256, 0, stream>>>(elog, (unsigned*)segm, dst, E);
    seg_expsum_k<<<gb(E), 256, 0, stream>>>(elog, (const unsigned*)segm, dst, ealpha, segs, E);
    edge_alpha_k<<<gb(E), 256, 0, stream>>>(ealpha, segs, dst, E);
    zero(t32, (long long)N * BD);
    scatter_add_k<<<gb((long long)E * BD), 256, 0, stream>>>(vb, src, dst, ealpha, t32, E, BD);
    gemm(hin, P.sw, P.sb, t32, BD, cat + 2 * BD, HD, N, HD, BD, 1);
    // gate + residual + LN(gelu)
    gemm(cat, P.gw, P.gb, nullptr, 0, gatebuf, HD, N, HD, HD, 2);
    gate_combine_k<<<gb(NH), 256, 0, stream>>>(gatebuf, cat, hin, comb, NH);
    lnorm(comb, P.lng, P.lnb, hout, 1);
    float* t = hin; hin = hout; hout = t;
  }

  // ---- temporal layers ----
  for (int ti = 0; ti < 2; ++ti) {
    const Tmp& T = tp[ti];
    float *xt = nbuf[0], *xtf = nbuf[1], *xln = nbuf[2];
    tscale_k<<<gb(NH), 256, 0, stream>>>(hin, T.scale, T.shift, xt, NH);
    gemm(xt, T.w1, T.b1, nullptr, 0, nb192, 2 * HD, N, HD, 2 * HD, 1);
    gemm(nb192, T.w2, T.b2, xt, HD, xtf, HD, N, 2 * HD, HD, 0);  // xt + ff
    lnorm(xtf, T.lng, T.lnb, xln, 0);
    tupdate_k<<<gb(NH), 256, 0, stream>>>(hin, xln, T.gate, NH);
  }

  // ---- GAT (4 heads, with self loops) ----
  {
    float *xp = nbuf[0], *go = nbuf[1];
    gemm(hin, gat_w, nullptr, nullptr, 0, xp, HD, N, HD, HD, 0);
    gat_coef_k<<<gb((long long)N * 4), 256, 0, stream>>>(xp, gat_as, gat_ad, asrc, adst, N);
    zero(mkey4, (long long)N * 4); zero(s4, (long long)N * 4);
    gat_logit_max_k<<<gb((long long)E * 4), 256, 0, stream>>>(asrc, adst, src, dst, lgE, (unsigned*)mkey4, E);
    gat_logit_max_k<<<gb((long long)N * 4), 256, 0, stream>>>(asrc, adst, nullptr, nullptr, lgN, (unsigned*)mkey4, N);
    gat_expsum_k<<<gb((long long)E * 4), 256, 0, stream>>>(lgE, (const unsigned*)mkey4, dst, s4, E);
    gat_expsum_k<<<gb((long long)N * 4), 256, 0, stream>>>(lgN, (const unsigned*)mkey4, nullptr, s4, N);
    gat_alpha_k<<<gb((long long)E * 4), 256, 0, stream>>>(lgE, s4, dst, E);
    gat_alpha_k<<<gb((long long)N * 4), 256, 0, stream>>>(lgN, s4, nullptr, N);
    zero(go, NH);
    gat_scatter_k<<<gb((long long)E * HD), 256, 0, stream>>>(xp, src, dst, lgE, go, E);
    gat_scatter_k<<<gb((long long)N * HD), 256, 0, stream>>>(xp, nullptr, nullptr, lgN, go, N);
    bn_gelu_add_k<<<gb(NH), 256, 0, stream>>>(go, gat_bias, bng, bnb, bnm, bnv, hin, NH);
  }

  // ---- feature branch: feat = gelu(ln(gelu(raw@w1+b1))@w2+b2) ----
  float* feat = nbuf[5];
  gemm(X, fw1, fb1, nullptr, 0, nbuf[0], HD, N, ICD, HD, 1);
  lnorm(nbuf[0], flng, flnb, nbuf[2], 0);
  gemm(nbuf[2], fw2, fb2, nullptr, 0, feat, HD, N, HD, HD, 1);

  // ---- heads ----
  gemm(hin, g1w, g1b, nullptr, 0, nbuf[0], HD, N, HD, HD, 1);
  gemm(nbuf[0], g2w, g2b, nullptr, 0, h48a, 48, N, HD, 48, 1);
  gemm(h48a, g3w, g3b, nullptr, 0, gp, OCD, N, 48, OCD, 0);
  gemm(feat, f1w, f1b, nullptr, 0, h48b, 48, N, HD, 48, 1);
  gemm(h48b, f2w, f2b, nullptr, 0, fp, OCD, N, 48, OCD, 0);
  // mix: [x|feat] @ w1 split into two GEMMs (w1 rows 0..95 / 96..191)
  gemm(hin, m1w, nullptr, nullptr, 0, nbuf[1], HD, N, HD, HD, 0);
  gemm(feat, m1w + HD * HD, m1b, nbuf[1], HD, nbuf[2], HD, N, HD, HD, 1);
  gemm(nbuf[2], m2w, m2b, nullptr, 0, mixv, OCD, N, HD, OCD, 2);
  gemm(X, skw, skb, nullptr, 0, skv, OCD, N, ICD, OCD, 0);
  final_mix_k<<<gb((long long)N * OCD), 256, 0, stream>>>(mixv, gp, fp, skv,
                                                          (float*)d_out,
                                                          (long long)N * OCD);
}